// InterChannelUpRaw_29180007809618
// MI455X (gfx1250) — compile-verified
//
#include <hip/hip_runtime.h>
#include <math.h>

// ---------------------------------------------------------------------------
// Problem constants (reference: n=8, c=256, h=w=192, K=6, stride 2, pad 2)
// ---------------------------------------------------------------------------
#define N_S    8
#define C_CH   256
#define H_IN   192
#define W_IN   192
#define H_OUT  96
#define W_OUT  96
#define PLANE  (H_IN * W_IN)       // 36864
#define OPLANE (H_OUT * W_OUT)     // 9216
#define GN_CNT (8 * OPLANE)        // elements per GroupNorm group = 73728

// Conv input staging tile: 8 output rows -> 20 input rows.
// Column layout: lds col = input col + 4  (left pad 4 so payload is 16B aligned)
//   cols 0..3   : zero halo (input cols -4..-1; only -2,-1 ever read)
//   cols 4..195 : input cols 0..191 (async b128 target, 16B aligned)
//   cols 196..199: zero halo (input cols 192,193)
//   cols 200..203: pad (never read)
#define T_ROWS 20
#define T_PITCH 204

#if defined(__gfx1250__) && __has_builtin(__builtin_amdgcn_s_wait_asynccnt)
  #if __has_builtin(__builtin_amdgcn_global_load_async_to_lds_b128)
    #define USE_ASYNC128 1
    #define USE_ASYNC32  0
  #elif __has_builtin(__builtin_amdgcn_global_load_async_to_lds_b32)
    #define USE_ASYNC128 0
    #define USE_ASYNC32  1
  #else
    #define USE_ASYNC128 0
    #define USE_ASYNC32  0
  #endif
#else
  #define USE_ASYNC128 0
  #define USE_ASYNC32  0
#endif

typedef __attribute__((address_space(1))) int gas_int;   // global dword addr
typedef __attribute__((address_space(3))) int las_int;   // LDS dword addr

typedef int v4i __attribute__((vector_size(16)));        // 4 x i32 (b128 payload)
typedef __attribute__((address_space(1))) v4i gas_v4i;   // global 4-dword addr
typedef __attribute__((address_space(3))) v4i las_v4i;   // LDS 4-dword addr

typedef __attribute__((ext_vector_type(2))) float v2f;
typedef __attribute__((ext_vector_type(8))) float v8f;

// ---------------------------------------------------------------------------
// One-time zeroing of the static halo columns (cols 0..3 and 196..199).
// ---------------------------------------------------------------------------
__device__ __forceinline__ void zero_halo(float* __restrict__ tile, int tid) {
    for (int i = tid; i < T_ROWS * 8; i += 256) {
        const int r = i >> 3, q = i & 7;
        const int col = (q < 4) ? q : (192 + q);
        tile[r * T_PITCH + col] = 0.0f;
    }
}

// ---------------------------------------------------------------------------
// Stage conv input strip (input rows 2*oy0-2 .. 2*oy0+17) into LDS via async
// b128 copies; out-of-range rows are zero-filled inline. Two barriers per fill.
// ---------------------------------------------------------------------------
__device__ __forceinline__ void fill_tile(float* __restrict__ tile,
                                          const float* __restrict__ plane,
                                          int oy0, int tid) {
    __syncthreads();  // previous consumers of tile are done
    for (int task = tid; task < T_ROWS * 48; task += 256) {   // 48 float4 / row
        const int r = task / 48;
        const int q = task - r * 48;
        const int iy = 2 * oy0 - 2 + r;
        float* dst = tile + r * T_PITCH + 4 + 4 * q;          // 16B aligned
        if (iy >= 0 && iy < H_IN) {
            const float* src = plane + (size_t)iy * W_IN + 4 * q;
#if USE_ASYNC128
            __builtin_amdgcn_global_load_async_to_lds_b128(
                (gas_v4i*)src, (las_v4i*)dst, 0, 0);
#elif USE_ASYNC32
            __builtin_amdgcn_global_load_async_to_lds_b32((gas_int*)(src + 0), (las_int*)(dst + 0), 0, 0);
            __builtin_amdgcn_global_load_async_to_lds_b32((gas_int*)(src + 1), (las_int*)(dst + 1), 0, 0);
            __builtin_amdgcn_global_load_async_to_lds_b32((gas_int*)(src + 2), (las_int*)(dst + 2), 0, 0);
            __builtin_amdgcn_global_load_async_to_lds_b32((gas_int*)(src + 3), (las_int*)(dst + 3), 0, 0);
#else
            *(float4*)dst = *(const float4*)src;
#endif
        } else {
            const float4 z = make_float4(0.f, 0.f, 0.f, 0.f);
            *(float4*)dst = z;
        }
    }
#if USE_ASYNC128 || USE_ASYNC32
    __builtin_amdgcn_s_wait_asynccnt(0);
#endif
    __syncthreads();
}

// 6x6 stride-2 conv at (l, ox) within the staged strip for both kernels.
// lds col = (2*ox + kx - 2) + 4 = 2*ox + kx + 2
__device__ __forceinline__ void conv2(const float* __restrict__ tile,
                                      const float* __restrict__ ku,
                                      const float* __restrict__ kv,
                                      int l, int ox, float& o1, float& o2) {
    float a1 = 0.f, a2 = 0.f;
#pragma unroll
    for (int ky = 0; ky < 6; ++ky) {
        const float* trow = tile + (2 * l + ky) * T_PITCH + 2 * ox + 2;
#pragma unroll
        for (int kx = 0; kx < 6; ++kx) {
            float xv = trow[kx];
            a1 = fmaf(ku[ky * 6 + kx], xv, a1);
            a2 = fmaf(kv[ky * 6 + kx], xv, a2);
        }
    }
    o1 = a1; o2 = a2;
}

// ---------------------------------------------------------------------------
// K1: per-(n,c) plane -> 36 block means (6x6 adaptive pool) + plane mean.
// 288 threads: 36 bins x 8 partial lanes, fixed-order reduction.
// ---------------------------------------------------------------------------
__global__ __launch_bounds__(288) void pool_kernel(const float* __restrict__ x,
                                                   float* __restrict__ pooled6,
                                                   float* __restrict__ meanbuf) {
    __shared__ float red[288];
    __shared__ float bins[36];
    const int tid = threadIdx.x;
    const int nc  = blockIdx.x;
    const float4* plane = (const float4*)(x + (size_t)nc * PLANE);
    const int bin = tid >> 3, sub = tid & 7;
    const int by = bin / 6, bx = bin - by * 6;
    float s = 0.f;
#pragma unroll
    for (int rr = 0; rr < 4; ++rr) {
        int y = by * 32 + sub * 4 + rr;
        const float4* p = plane + (size_t)y * 48 + bx * 8;
#pragma unroll
        for (int i = 0; i < 8; ++i) {
            float4 v = p[i];
            s += v.x + v.y + v.z + v.w;
        }
    }
    red[tid] = s;
    __syncthreads();
    if (sub == 0) {
        float b = 0.f;
#pragma unroll
        for (int i = 0; i < 8; ++i) b += red[tid + i];
        bins[bin] = b;
    }
    __syncthreads();
    if (tid < 36) pooled6[(size_t)nc * 36 + tid] = bins[tid] * (1.0f / 1024.0f);
    if (tid == 0) {
        float t = 0.f;
        for (int i = 0; i < 36; ++i) t += bins[i];
        meanbuf[nc] = t * (1.0f / 36864.0f);
    }
}

// ---------------------------------------------------------------------------
// K2: projection GEMMs via V_WMMA_F32_16X16X4_F32 (single wave per block).
//   waves 0..17 : C(288x16) = pooled6^T(288x256) x B(256x16), B cols {w_u,w_v}
//   waves 18..20: C(48x16)  = w_up(36x256 pad48) x means(256x8 pad16)
// All loads are in-bounds (clamped addresses); masking is arithmetic, so the
// K-loop is branch-free: load -> select -> v_wmma.
// ---------------------------------------------------------------------------
__global__ __launch_bounds__(32) void proj_kernel(const float* __restrict__ pooled6,
                                                  const float* __restrict__ meanbuf,
                                                  const float* __restrict__ w_u,
                                                  const float* __restrict__ b_u,
                                                  const float* __restrict__ w_v,
                                                  const float* __restrict__ b_v,
                                                  const float* __restrict__ w_up,
                                                  const float* __restrict__ b_up,
                                                  float* __restrict__ ubuf,
                                                  float* __restrict__ vbuf,
                                                  float* __restrict__ pubuf) {
    const int wave = blockIdx.x;
    const int lane = threadIdx.x;
    const int half = lane >> 4;   // K sub-block select
    const int l16  = lane & 15;   // N column / M row index
    v8f acc = {0.f, 0.f, 0.f, 0.f, 0.f, 0.f, 0.f, 0.f};

    if (wave < 18) {
        // M row = n*36 + t, exactly 288 rows = 18 tiles (no padding)
        const int m = wave * 16 + l16;
        const int n = m / 36, t = m - n * 36;
        const float mU = (l16 == 0) ? 1.f : 0.f;
        const float mV = (l16 == 1) ? 1.f : 0.f;
        const float* Ab = pooled6 + (size_t)n * C_CH * 36 + t;
        for (int k0 = 0; k0 < C_CH; k0 += 4) {
            const int c0 = k0 + 2 * half;
            v2f a, bm;
            a.x = Ab[(size_t)c0 * 36];
            a.y = Ab[(size_t)(c0 + 1) * 36];
            bm.x = w_u[c0] * mU + w_v[c0] * mV;
            bm.y = w_u[c0 + 1] * mU + w_v[c0 + 1] * mV;
            acc = __builtin_amdgcn_wmma_f32_16x16x4_f32(false, a, false, bm,
                                                        (short)0, acc, false, false);
        }
        const float bu = b_u[0], bv = b_v[0];
#pragma unroll
        for (int r = 0; r < 8; ++r) {
            const int mg = wave * 16 + r + 8 * half;
            const int nn = mg / 36, tt = mg - nn * 36;
            if (l16 == 0)      ubuf[nn * 36 + tt] = acc[r] + bu;
            else if (l16 == 1) vbuf[nn * 36 + tt] = acc[r] + bv;
        }
    } else {
        const int pt = wave - 18;          // 0..2
        const int m  = pt * 16 + l16;      // row of w_up (pad to 48)
        const float mA = (m < 36) ? 1.f : 0.f;
        const float mB = (l16 < 8) ? 1.f : 0.f;
        const float* Ab = w_up + (size_t)((m < 36) ? m : 0) * C_CH;      // clamped
        const float* Bb = meanbuf + (size_t)((l16 < 8) ? l16 : 0) * C_CH;
        for (int k0 = 0; k0 < C_CH; k0 += 4) {
            const int c0 = k0 + 2 * half;
            v2f a, bm;
            a.x = Ab[c0] * mA;
            a.y = Ab[c0 + 1] * mA;
            bm.x = Bb[c0] * mB;
            bm.y = Bb[c0 + 1] * mB;
            acc = __builtin_amdgcn_wmma_f32_16x16x4_f32(false, a, false, bm,
                                                        (short)0, acc, false, false);
        }
#pragma unroll
        for (int r = 0; r < 8; ++r) {
            const int k = pt * 16 + r + 8 * half;
            if (k < 36 && l16 < 8) pubuf[l16 * 36 + k] = acc[r] + b_up[k];
        }
    }
}

// ---------------------------------------------------------------------------
// K3: GroupNorm statistics. One block per (n, group): 8 channels x 96x96
// conv outputs, accumulate sum/sumsq for both conv tensors.
// ---------------------------------------------------------------------------
__global__ __launch_bounds__(256) void stats_kernel(const float* __restrict__ x,
                                                    const float* __restrict__ ubuf,
                                                    const float* __restrict__ vbuf,
                                                    float* __restrict__ gstats) {
    __shared__ float tile[T_ROWS * T_PITCH];
    __shared__ float ku[36], kv[36];
    __shared__ float red[256];
    const int tid = threadIdx.x;
    const int n = blockIdx.x >> 5;
    const int g = blockIdx.x & 31;
    if (tid < 36) { ku[tid] = ubuf[n * 36 + tid]; kv[tid] = vbuf[n * 36 + tid]; }
    zero_halo(tile, tid);
    float s1 = 0.f, q1 = 0.f, s2 = 0.f, q2 = 0.f;
    for (int ci = 0; ci < 8; ++ci) {
        const float* plane = x + (size_t)(n * C_CH + g * 8 + ci) * PLANE;
        for (int strip = 0; strip < 12; ++strip) {
            fill_tile(tile, plane, strip * 8, tid);  // barriers cover ku/halo too
#pragma unroll
            for (int j = 0; j < 3; ++j) {
                const int idx = tid + j * 256;         // 0..767 in 8x96 strip
                const int l = idx / 96, ox = idx - l * 96;
                float o1, o2;
                conv2(tile, ku, kv, l, ox, o1, o2);
                s1 += o1; q1 += o1 * o1;
                s2 += o2; q2 += o2 * o2;
            }
        }
    }
    float vals[4] = {s1, q1, s2, q2};
#pragma unroll
    for (int k = 0; k < 4; ++k) {
        __syncthreads();
        red[tid] = vals[k];
        __syncthreads();
        for (int off = 128; off > 0; off >>= 1) {
            if (tid < off) red[tid] += red[tid + off];
            __syncthreads();
        }
        if (tid == 0) gstats[blockIdx.x * 4 + k] = red[0];
    }
}

// ---------------------------------------------------------------------------
// K4: recompute convs, apply folded GroupNorm + leaky, accumulate channel-
// partial product. Block = (n, 8-row strip, 32-channel group) -> 768 blocks.
// ---------------------------------------------------------------------------
__global__ __launch_bounds__(256) void prod_kernel(const float* __restrict__ x,
                                                   const float* __restrict__ ubuf,
                                                   const float* __restrict__ vbuf,
                                                   const float* __restrict__ gstats,
                                                   const float* __restrict__ gamma_u,
                                                   const float* __restrict__ beta_u,
                                                   const float* __restrict__ gamma_v,
                                                   const float* __restrict__ beta_v,
                                                   float* __restrict__ partial) {
    __shared__ float tile[T_ROWS * T_PITCH];
    __shared__ float ku[36], kv[36];
    const int tid = threadIdx.x;
    const int b = blockIdx.x;
    const int n = b / 96;
    const int rem = b - n * 96;
    const int strip = rem >> 3;
    const int cg = rem & 7;
    if (tid < 36) { ku[tid] = ubuf[n * 36 + tid]; kv[tid] = vbuf[n * 36 + tid]; }
    zero_halo(tile, tid);
    float acc0 = 0.f, acc1 = 0.f, acc2 = 0.f;
    const float inv_cnt = 1.0f / (float)GN_CNT;
    for (int ci = 0; ci < 32; ++ci) {
        const int c = cg * 32 + ci;
        const int g = c >> 3;
        const float* gs = gstats + (n * 32 + g) * 4;
        float m1 = gs[0] * inv_cnt;
        float v1 = fmaxf(gs[1] * inv_cnt - m1 * m1, 0.f);
        float a1 = rsqrtf(v1 + 1e-5f) * gamma_u[c];
        float c1 = beta_u[c] - m1 * a1;
        float m2 = gs[2] * inv_cnt;
        float v2 = fmaxf(gs[3] * inv_cnt - m2 * m2, 0.f);
        float a2 = rsqrtf(v2 + 1e-5f) * gamma_v[c];
        float c2 = beta_v[c] - m2 * a2;
        const float* plane = x + (size_t)(n * C_CH + c) * PLANE;
        fill_tile(tile, plane, strip * 8, tid);
#pragma unroll
        for (int j = 0; j < 3; ++j) {
            const int idx = tid + j * 256;
            const int l = idx / 96, ox = idx - l * 96;
            float o1, o2;
            conv2(tile, ku, kv, l, ox, o1, o2);
            float y1 = fmaf(o1, a1, c1);
            y1 = (y1 >= 0.f) ? y1 : 0.01f * y1;
            float y2 = fmaf(o2, a2, c2);
            y2 = (y2 >= 0.f) ? y2 : 0.01f * y2;
            float pr = y1 * y2;
            if (j == 0) acc0 += pr; else if (j == 1) acc1 += pr; else acc2 += pr;
        }
    }
    const size_t base = (size_t)(n * 8 + cg) * OPLANE + strip * 768;
    partial[base + tid]       = acc0;
    partial[base + tid + 256] = acc1;
    partial[base + tid + 512] = acc2;
}

// ---------------------------------------------------------------------------
// K5: sum 8 channel partials, /sqrt(c), outer product with p_u, pixel-shuffle
// indexing, sigmoid then softmax over the 9 shuffle groups.
// ---------------------------------------------------------------------------
__global__ __launch_bounds__(256) void final_kernel(const float* __restrict__ partial,
                                                    const float* __restrict__ pubuf,
                                                    float* __restrict__ out) {
    __shared__ float pu[36];
    const int tid = threadIdx.x;
    const int gid = blockIdx.x * 256 + tid;      // 0..73727, n uniform per block
    const int n = gid / OPLANE;
    const int p = gid - n * OPLANE;
    if (tid < 36) pu[tid] = pubuf[n * 36 + tid];
    __syncthreads();
    float s = 0.f;
#pragma unroll
    for (int cg = 0; cg < 8; ++cg) s += partial[(size_t)(n * 8 + cg) * OPLANE + p];
    s *= 0.0625f;                                 // 1/sqrt(256)
    const int y = p / 96, xq = p - y * 96;
#pragma unroll
    for (int off = 0; off < 4; ++off) {
        const int dy = off >> 1, dx = off & 1;
        float e[9];
        float mx = -1e30f;
#pragma unroll
        for (int j = 0; j < 9; ++j) {
            float tv = pu[4 * j + off] * s;       // t = j*4 + dy*2 + dx
            float sg = 1.0f / (1.0f + __expf(-tv));
            e[j] = sg;
            mx = fmaxf(mx, sg);
        }
        float sum = 0.f;
#pragma unroll
        for (int j = 0; j < 9; ++j) { e[j] = __expf(e[j] - mx); sum += e[j]; }
        const float inv = 1.0f / sum;
        const int yy = 2 * y + dy, xx = 2 * xq + dx;
#pragma unroll
        for (int j = 0; j < 9; ++j)
            out[(((size_t)n * 9 + j) * H_IN + yy) * W_IN + xx] = e[j] * inv;
    }
}

// ---------------------------------------------------------------------------
extern "C" void kernel_launch(void* const* d_in, const int* in_sizes, int n_in,
                              void* d_out, int out_size, void* d_ws, size_t ws_size,
                              hipStream_t stream) {
    const float* x       = (const float*)d_in[0];
    const float* w_u     = (const float*)d_in[1];
    const float* b_u     = (const float*)d_in[2];
    const float* w_v     = (const float*)d_in[3];
    const float* b_v     = (const float*)d_in[4];
    const float* w_up    = (const float*)d_in[5];
    const float* b_up    = (const float*)d_in[6];
    const float* gamma_u = (const float*)d_in[7];
    const float* beta_u  = (const float*)d_in[8];
    const float* gamma_v = (const float*)d_in[9];
    const float* beta_v  = (const float*)d_in[10];
    float* out = (float*)d_out;

    float* ws      = (float*)d_ws;
    float* pooled6 = ws;                              // 8*256*36 = 73728
    float* meanbuf = pooled6 + N_S * C_CH * 36;       // 8*256    = 2048
    float* ubuf    = meanbuf + N_S * C_CH;            // 8*36
    float* vbuf    = ubuf + N_S * 36;                 // 8*36
    float* pubuf   = vbuf + N_S * 36;                 // 8*36
    float* gstats  = pubuf + N_S * 36;                // 8*32*4   = 1024
    float* partial = gstats + N_S * 32 * 4;           // 8*8*9216 = 589824
    // total ~2.7 MB of f32 workspace

    pool_kernel<<<N_S * C_CH, 288, 0, stream>>>(x, pooled6, meanbuf);
    proj_kernel<<<21, 32, 0, stream>>>(pooled6, meanbuf, w_u, b_u, w_v, b_v,
                                       w_up, b_up, ubuf, vbuf, pubuf);
    stats_kernel<<<N_S * 32, 256, 0, stream>>>(x, ubuf, vbuf, gstats);
    prod_kernel<<<N_S * 96, 256, 0, stream>>>(x, ubuf, vbuf, gstats,
                                              gamma_u, beta_u, gamma_v, beta_v,
                                              partial);
    final_kernel<<<(N_S * OPLANE) / 256, 256, 0, stream>>>(partial, pubuf, out);
}